// Recommender_79508434584054
// MI455X (gfx1250) — compile-verified
//
#include <hip/hip_runtime.h>

#define N_USERS    50000
#define N_ENTITIES 100000
#define N_EDGES    2000000
#define NNZ        1000000
#define N_REL      17
#define DIM        64
#define N_HOPS     3

typedef float v2f __attribute__((ext_vector_type(2)));
typedef float v8f __attribute__((ext_vector_type(8)));

static inline int cdiv_i(long a, long b) { return (int)((a + b - 1) / b); }

// res <- emb (both user and entity regions of d_out), ent_cur <- entity_emb
__global__ __launch_bounds__(256) void rec_init_kernel(
    const float* __restrict__ user_emb,
    const float* __restrict__ entity_emb,
    float* __restrict__ out,       // [NE*DIM entity_res | NU*DIM user_res]
    float* __restrict__ ent_cur) {
  long i = (long)blockIdx.x * 256 + threadIdx.x;
  const long ne = (long)N_ENTITIES * DIM;
  const long nu = (long)N_USERS * DIM;
  if (i < ne) {
    float v = entity_emb[i];
    out[i] = v;
    ent_cur[i] = v;
  } else if (i < ne + nu) {
    out[i] = user_emb[i - ne];
  }
}

__global__ __launch_bounds__(256) void rec_zero_kernel(
    float* __restrict__ ent_agg, float* __restrict__ usr_agg) {
  long i = (long)blockIdx.x * 256 + threadIdx.x;
  const long ne = (long)N_ENTITIES * DIM;
  const long nu = (long)N_USERS * DIM;
  if (i < ne)            ent_agg[i] = 0.0f;
  else if (i < ne + nu)  usr_agg[i - ne] = 0.0f;
}

// KG aggregate: ent_agg[head] += ent[tail] * unmask * weight[etype-1]
// 16 lanes per edge, float4 per lane; weight table staged in LDS.
__global__ __launch_bounds__(256) void rec_edge_kernel(
    const float* __restrict__ ent,      // [NE, DIM]
    const float* __restrict__ weight,   // [N_REL-1, DIM]
    const float* __restrict__ unmask,   // [E]
    const int*   __restrict__ head,     // edge_index[0]
    const int*   __restrict__ tail,     // edge_index[1]
    const int*   __restrict__ etype,    // [E], values 1..16
    float* __restrict__ ent_agg) {
  __shared__ float wlds[(N_REL - 1) * DIM];   // 4 KB
  for (int i = threadIdx.x; i < (N_REL - 1) * DIM; i += 256)
    wlds[i] = weight[i];
  __syncthreads();

  long gid = (long)blockIdx.x * 256 + threadIdx.x;
  long e = gid >> 4;
  int  q = (int)(gid & 15);
  if (e >= N_EDGES) return;

  int   h = head[e];
  int   t = tail[e];
  int   r = etype[e] - 1;
  float u = unmask[e];

  float4 x = *(const float4*)(ent  + (long)t * DIM + q * 4);
  float4 w = *(const float4*)(wlds +        r * DIM + q * 4);
  float* dst = ent_agg + (long)h * DIM + q * 4;
  atomicAdd(dst + 0, x.x * u * w.x);
  atomicAdd(dst + 1, x.y * u * w.y);
  atomicAdd(dst + 2, x.z * u * w.z);
  atomicAdd(dst + 3, x.w * u * w.w);
}

// User aggregate: usr_agg[row] += ent[col]; 16 lanes per nnz, float4 per lane.
__global__ __launch_bounds__(256) void rec_user_kernel(
    const float* __restrict__ ent,
    const int*   __restrict__ rows,
    const int*   __restrict__ cols,
    float* __restrict__ usr_agg) {
  long gid = (long)blockIdx.x * 256 + threadIdx.x;
  long n = gid >> 4;
  int  q = (int)(gid & 15);
  if (n >= NNZ) return;

  int rr = rows[n];
  int cc = cols[n];
  float4 x = *(const float4*)(ent + (long)cc * DIM + q * 4);
  float* dst = usr_agg + (long)rr * DIM + q * 4;
  atomicAdd(dst + 0, x.x);
  atomicAdd(dst + 1, x.y);
  atomicAdd(dst + 2, x.z);
  atomicAdd(dst + 3, x.w);
}

// L2-normalize 16-row tiles and accumulate into res (and optionally next).
// Row squared-norms computed as diag(X * X^T) via 16x V_WMMA_F32_16X16X4_F32.
// For the symmetric Gram product the A(16x4) and B(4x16) per-lane layouts
// coincide: lanes 0-15 hold row=lane, K={4k,4k+1}; lanes 16-31 hold
// row=lane-16, K={4k+2,4k+3}. Each wave loads its 16x64 tile exactly once.
__global__ __launch_bounds__(256) void rec_normalize_kernel(
    const float* __restrict__ agg,   // [nrows, DIM]
    float* __restrict__ res,         // [nrows, DIM], accumulated (+=)
    float* __restrict__ next,        // [nrows, DIM] or nullptr
    int nrows) {
  int wave = (int)(((long)blockIdx.x * 256 + threadIdx.x) >> 5);
  int lane = threadIdx.x & 31;
  int row0 = wave * 16;
  if (row0 >= nrows) return;           // wave-uniform; EXEC all-1s for WMMA

  int half = lane >> 4;                // which column pair of the K-quad
  int m    = lane & 15;                // row within the 16-row tile
  long base = (long)(row0 + m) * DIM + half * 2;

  float2 vals[16];
#pragma unroll
  for (int k = 0; k < 16; ++k)
    vals[k] = *(const float2*)(agg + base + 4 * k);

  v8f c = {};
#pragma unroll
  for (int k = 0; k < 16; ++k) {
    v2f a;
    a[0] = vals[k].x;
    a[1] = vals[k].y;
    // D(16x16) += A(16x4) * B(4x16); A and B fed from the same registers.
    c = __builtin_amdgcn_wmma_f32_16x16x4_f32(
        /*neg_a=*/false, a, /*neg_b=*/false, a,
        /*c_mod=*/(short)0, c, /*reuse_a=*/false, /*reuse_b=*/false);
  }

  // Extract Gram diagonal: row mm<8 -> c[mm]@lane mm ; mm>=8 -> c[mm-8]@lane mm+16
  int idx = (lane < 8) ? lane : ((lane - 24) & 7);
  float diag = 0.0f;
#pragma unroll
  for (int v = 0; v < 8; ++v)
    if (idx == v) diag = c[v];

  int src = (m < 8) ? m : (m + 16);
  float nsq = __shfl(diag, src, 32);
  float inv = 1.0f / fmaxf(sqrtf(nsq), 1e-12f);

#pragma unroll
  for (int k = 0; k < 16; ++k) {
    float a0 = vals[k].x * inv;
    float a1 = vals[k].y * inv;
    long  o  = base + 4 * k;
    res[o]     += a0;
    res[o + 1] += a1;
    if (next) {
      next[o]     = a0;
      next[o + 1] = a1;
    }
  }
}

extern "C" void kernel_launch(void* const* d_in, const int* in_sizes, int n_in,
                              void* d_out, int out_size, void* d_ws, size_t ws_size,
                              hipStream_t stream) {
  const float* user_emb   = (const float*)d_in[0];   // [NU, DIM]
  const float* entity_emb = (const float*)d_in[1];   // [NE, DIM]
  const float* weight     = (const float*)d_in[2];   // [N_REL-1, DIM]
  const float* unmask     = (const float*)d_in[3];   // [E]
  const int*   edge_index = (const int*)d_in[4];     // [2, E]
  const int*   edge_type  = (const int*)d_in[5];     // [E]
  const int*   irows      = (const int*)d_in[6];     // [NNZ]
  const int*   icols      = (const int*)d_in[7];     // [NNZ]
  (void)in_sizes; (void)n_in; (void)out_size; (void)ws_size;

  const int* head = edge_index;
  const int* tail = edge_index + N_EDGES;

  float* out     = (float*)d_out;                        // [NE*DIM | NU*DIM]
  float* ent_cur = (float*)d_ws;                         // 25.6 MB
  float* ent_agg = ent_cur + (long)N_ENTITIES * DIM;     // 25.6 MB
  float* usr_agg = ent_agg + (long)N_ENTITIES * DIM;     // 12.8 MB

  const long totElems = (long)(N_ENTITIES + N_USERS) * DIM;
  const int gInit = cdiv_i(totElems, 256);
  const int gEdge = cdiv_i((long)N_EDGES * 16, 256);
  const int gUser = cdiv_i((long)NNZ * 16, 256);
  const int gNrmE = cdiv_i(((long)N_ENTITIES / 16) * 32, 256);
  const int gNrmU = cdiv_i(((long)N_USERS / 16) * 32, 256);

  rec_init_kernel<<<gInit, 256, 0, stream>>>(user_emb, entity_emb, out, ent_cur);

  for (int hop = 0; hop < N_HOPS; ++hop) {
    rec_zero_kernel<<<gInit, 256, 0, stream>>>(ent_agg, usr_agg);
    rec_edge_kernel<<<gEdge, 256, 0, stream>>>(ent_cur, weight, unmask,
                                               head, tail, edge_type, ent_agg);
    rec_user_kernel<<<gUser, 256, 0, stream>>>(ent_cur, irows, icols, usr_agg);
    // users first (reads usr_agg only), then entities (overwrites ent_cur)
    rec_normalize_kernel<<<gNrmU, 256, 0, stream>>>(
        usr_agg, out + (long)N_ENTITIES * DIM, nullptr, N_USERS);
    rec_normalize_kernel<<<gNrmE, 256, 0, stream>>>(
        ent_agg, out, ent_cur, N_ENTITIES);
  }
}